// E60bGatedResidualCell_45792941310472
// MI455X (gfx1250) — compile-verified
//
#include <hip/hip_runtime.h>
#include <math.h>

typedef __attribute__((ext_vector_type(2))) float v2f;
typedef __attribute__((ext_vector_type(8))) float v8f;

#define T_STEPS 512
#define BATCH   32
#define DIM     1024
#define NBLK    64          // blocks in the persistent scan grid (one per e-tile)

// ---------------------------------------------------------------------------
// D = A(16x4) * B(4x16) + C  f32 WMMA fragment helpers (wave32).
// A layout: lane l: M = l&15, holds K = (l>>4)*2 + {0,1} in {v0,v1}
// B layout: lane l: N = l&15, holds K = (l>>4)*2 + {0,1} in {v0,v1}
// C/D layout: lane l, vgpr v: M = v + (l>>4)*8, N = l&15
// ---------------------------------------------------------------------------

__device__ __forceinline__ v8f wmma_f32(v2f a, v2f b, v8f c) {
    return __builtin_amdgcn_wmma_f32_16x16x4_f32(
        /*neg_a=*/false, a, /*neg_b=*/false, b,
        /*c_mod=*/(short)0, c, /*reuse_a=*/false, /*reuse_b=*/false);
}

__device__ __forceinline__ float sigmoidf_(float x) {
    return 1.0f / (1.0f + __expf(-x));
}

// ---------------------------------------------------------------------------
// Kernel 0: zero h[0] (B*D floats) and the device barrier counter.
// ---------------------------------------------------------------------------
__global__ __launch_bounds__(256) void init_kernel(float* __restrict__ h_region,
                                                   unsigned* __restrict__ counter) {
    int i = blockIdx.x * 256 + threadIdx.x;
    if (i < BATCH * DIM) h_region[i] = 0.0f;
    if (i == 0) *counter = 0u;
}

// ---------------------------------------------------------------------------
// Kernel 1: precompute Wx_all and gate_all.
//   M = T*B = 16384 rows of x, N = DIM, K = DIM.
//   wx[mg, e]   = sum_d x[mg,d] * W_x[e,d]            -> out_region[mg*D + e]
//   gate[mg, e] = sigmoid(sum_d x[mg,d]*W_g[e,d]+b_g) -> h_region[(mg+B)*D + e]
// One wave per 16x16 tile, dual accumulators sharing the A (x) fragment.
// ---------------------------------------------------------------------------
__global__ __launch_bounds__(256) void precompute_kernel(
    const float* __restrict__ x,
    const float* __restrict__ Wx,
    const float* __restrict__ Wg,
    const float* __restrict__ bg,
    float* __restrict__ out_region,   // receives Wx_all (consumed+overwritten in scan)
    float* __restrict__ h_region)     // h[t+1] slots receive gate (consumed+overwritten)
{
    const int lane = threadIdx.x & 31;
    const int wave = (blockIdx.x << 3) | (threadIdx.x >> 5);  // 0..65535
    const int m0 = (wave >> 6) << 4;   // row tile over T*B
    const int e0 = (wave & 63) << 4;   // col tile over DIM

    const int mn = lane & 15;              // M for A, N for B
    const int kb = (lane >> 4) << 1;       // K sub-offset {0,2}

    const float* arow = x  + (size_t)(m0 + mn) * DIM + kb;
    const float* bxro = Wx + (size_t)(e0 + mn) * DIM + kb;
    const float* bgro = Wg + (size_t)(e0 + mn) * DIM + kb;

    v8f cx = {}; v8f cg = {};
#pragma unroll 4
    for (int k = 0; k < DIM; k += 4) {
        v2f a   = *reinterpret_cast<const v2f*>(arow + k);
        v2f bx  = *reinterpret_cast<const v2f*>(bxro + k);
        v2f bgv = *reinterpret_cast<const v2f*>(bgro + k);
        cx = wmma_f32(a, bx,  cx);
        cg = wmma_f32(a, bgv, cg);
    }

    const int n    = lane & 15;
    const int mhi  = (lane >> 4) << 3;
    const int e    = e0 + n;
    const float bge = bg[e];
#pragma unroll
    for (int v = 0; v < 8; ++v) {
        const int mg = m0 + mhi + v;                     // global row in [T*B]
        out_region[(size_t)mg * DIM + e] = cx[v];
        // (t+1)*B + b == mg + B  -> gate parked in the h[t+1] slot
        h_region[(size_t)(mg + BATCH) * DIM + e] = sigmoidf_(cg[v] + bge);
    }
}

// ---------------------------------------------------------------------------
// Kernel 2: persistent scan over all T steps.
//   pre[b,e] = sum_d h[t][b,d] * W_h[e,d] + wx[t][b,e] + bias[e]
//   h[t+1]   = h[t] + gate[t] * tanh(pre)
//   out[t]   = hn * hn * sigmoid(hn)
//
// One block per 32x16 e-tile (NBLK=64 blocks, 256 thr). K split 8 ways across
// the block's waves (dependent WMMA chain = 32/step). Each wave's W_h K-slice
// is invariant across t -> preloaded into 32 v2f register fragments before
// the time loop; the critical loop only streams h_prev. Steps are separated
// by a device-wide phase barrier on an atomic counter.
// ---------------------------------------------------------------------------
__global__ __launch_bounds__(256) void scan_kernel(
    const float* __restrict__ Wh,
    const float* __restrict__ bias,
    float* __restrict__ out_region,
    float* __restrict__ h_region,
    unsigned* __restrict__ counter)
{
    __shared__ float red[8 * 32 * 16];   // [kslice][b][n] = 16 KB

    const int tid  = threadIdx.x;
    const int lane = tid & 31;
    const int wv   = tid >> 5;           // 0..7 : K slice id
    const int e0   = blockIdx.x << 4;    // e-tile base

    const int mn = lane & 15;
    const int kb = (lane >> 4) << 1;
    const int k0 = wv << 7;              // 128-wide K slice

    // --- Preload this wave's W_h slice into registers (invariant over t) ---
    v2f bw[32];
    {
        const float* bwrow = Wh + (size_t)(e0 + mn) * DIM + k0 + kb;
#pragma unroll
        for (int i = 0; i < 32; ++i)
            bw[i] = *reinterpret_cast<const v2f*>(bwrow + 4 * i);
    }

    // Epilogue constants: both elements a thread handles share the same e.
    const int   eg = e0 + (tid & 15);
    const float be = bias[eg];
    const int   n   = lane & 15;
    const int   mhi = (lane >> 4) << 3;

    const size_t a0off = (size_t)mn * DIM + k0 + kb;          // rows 0..15
    const size_t a1off = (size_t)(16 + mn) * DIM + k0 + kb;   // rows 16..31

    for (int t = 0; t < T_STEPS; ++t) {
        const float* h_prev = h_region + (size_t)t * BATCH * DIM;
        float*       h_next = h_region + (size_t)(t + 1) * BATCH * DIM;
        float*       out_t  = out_region + (size_t)t * BATCH * DIM;

        v8f c0 = {}; v8f c1 = {};
        const float* a0p = h_prev + a0off;
        const float* a1p = h_prev + a1off;
#pragma unroll 4
        for (int i = 0; i < 32; ++i) {
            v2f a0 = *reinterpret_cast<const v2f*>(a0p + 4 * i);
            v2f a1 = *reinterpret_cast<const v2f*>(a1p + 4 * i);
            c0 = wmma_f32(a0, bw[i], c0);
            c1 = wmma_f32(a1, bw[i], c1);
        }

        // Park partial tiles in LDS in (b, n) layout.
        float* my = red + wv * 512;
#pragma unroll
        for (int v = 0; v < 8; ++v) {
            my[(mhi + v) * 16 + n]      = c0[v];
            my[(16 + mhi + v) * 16 + n] = c1[v];
        }
        __syncthreads();

        // 512 tile elements, 256 threads -> 2 each; 8-way reduction + epilogue.
#pragma unroll
        for (int r = 0; r < 2; ++r) {
            const int elem = tid + (r << 8);       // 0..511
            const int b    = elem >> 4;
            float s = 0.0f;
#pragma unroll
            for (int w = 0; w < 8; ++w) s += red[w * 512 + elem];

            const size_t idx = (size_t)b * DIM + eg;
            const float wx = out_t[idx];           // Wx_all parked here
            const float g  = h_next[idx];          // gate parked here
            const float hp = h_prev[idx];
            const float hn = hp + g * tanhf(s + wx + be);
            h_next[idx] = hn;
            out_t[idx]  = hn * hn * sigmoidf_(hn);
        }

        // ---- device-wide phase barrier between steps ----
        __threadfence();        // make this thread's h_next/out stores visible
        __syncthreads();        // whole block has fenced; LDS reads done too
        if (tid == 0) {
            __hip_atomic_fetch_add(counter, 1u, __ATOMIC_RELEASE,
                                   __HIP_MEMORY_SCOPE_AGENT);
            const unsigned target = (unsigned)NBLK * (unsigned)(t + 1);
            while (__hip_atomic_load(counter, __ATOMIC_ACQUIRE,
                                     __HIP_MEMORY_SCOPE_AGENT) < target) {
                __builtin_amdgcn_s_sleep(1);
            }
        }
        __syncthreads();        // release whole block into step t+1
    }
}

// ---------------------------------------------------------------------------
// Launch
// ---------------------------------------------------------------------------
extern "C" void kernel_launch(void* const* d_in, const int* in_sizes, int n_in,
                              void* d_out, int out_size, void* d_ws, size_t ws_size,
                              hipStream_t stream) {
    (void)in_sizes; (void)n_in; (void)out_size; (void)ws_size;

    const float* x   = (const float*)d_in[0];
    const float* W_h = (const float*)d_in[1];
    const float* W_x = (const float*)d_in[2];
    const float* W_g = (const float*)d_in[3];
    const float* b   = (const float*)d_in[4];
    const float* b_g = (const float*)d_in[5];

    float* out_region = (float*)d_out;                                   // [T,B,D]
    float* h_region   = (float*)d_out + (size_t)T_STEPS * BATCH * DIM;   // [T+1,B,D]
    unsigned* counter = (unsigned*)d_ws;

    // h[0] = 0, barrier counter = 0 (fresh every call -> graph-replay safe)
    init_kernel<<<(BATCH * DIM + 255) / 256, 256, 0, stream>>>(h_region, counter);

    // Wx_all -> out_region, gate -> h[t+1] slots
    precompute_kernel<<<8192, 256, 0, stream>>>(x, W_x, W_g, b_g,
                                                out_region, h_region);

    // Entire scan in one persistent kernel; inter-step sync via device barrier.
    scan_kernel<<<NBLK, 256, 0, stream>>>(W_h, b, out_region, h_region, counter);
}